// ProjectedAdaptiveLogSoftmax_59871844107158
// MI455X (gfx1250) — compile-verified
//
#include <hip/hip_runtime.h>
#include <math.h>

typedef float v2f __attribute__((ext_vector_type(2)));
typedef float v8f __attribute__((ext_vector_type(8)));

#define NEG_BIG    (-1.0e30f)
#define NEG_THRESH (-1.0e29f)

#define WMMA_F32(a, b, c) \
  __builtin_amdgcn_wmma_f32_16x16x4_f32(false, (a), false, (b), (short)0, (c), false, false)

__device__ __forceinline__ void lse_combine(float& M, float& S, float mm, float ss) {
  if (mm <= NEG_THRESH) return;            // empty / masked partial
  if (mm > M) { S = S * __expf(M - mm) + ss; M = mm; }
  else        { S += ss * __expf(mm - M); }
}

// ---------------------------------------------------------------------------
// Projection GEMM: C[N,V] = A[N,K] @ B[K,V]   (B row-major [K,V])
// Block: 256 threads = 8 waves; block tile 16 rows x 128 cols, wave tile 16x16.
// (Tiny fraction of total work; kept simple.)
// ---------------------------------------------------------------------------
__global__ __launch_bounds__(256) void wmma_proj_kernel(
    const float* __restrict__ A, const float* __restrict__ B,
    float* __restrict__ C, int N, int K, int V)
{
  const int row0 = blockIdx.y * 16;
  const int wave = threadIdx.x >> 5;
  const int lane = threadIdx.x & 31;
  const int half = lane >> 4;
  const int ln   = lane & 15;
  const int col0 = blockIdx.x * 128 + wave * 16;   // wave-uniform

  __shared__ float As[16][66];

  v8f c = {};
  for (int k0 = 0; k0 < K; k0 += 64) {
    const int kc = min(64, K - k0);
    __syncthreads();
    for (int i = threadIdx.x; i < 16 * 64; i += 256) {
      const int r = i >> 6, kl = i & 63;
      if (kl < kc) As[r][kl] = A[(size_t)(row0 + r) * K + k0 + kl];
    }
    __syncthreads();
    if (col0 < V) {                                // wave-uniform: EXEC stays full
#pragma unroll 4
      for (int kk = 0; kk < kc; kk += 4) {
        const int kg = kk + 2 * half;
        v2f a = *(const v2f*)(&As[ln][kg]);
        v2f b;
        b.x = B[(size_t)(k0 + kg) * V + col0 + ln];
        b.y = B[(size_t)(k0 + kg + 1) * V + col0 + ln];
        c = WMMA_F32(a, b, c);
      }
    }
  }
  if (col0 < V) {
    for (int r = 0; r < 8; ++r) {
      const int row = row0 + r + 8 * half;
      C[(size_t)row * V + col0 + ln] = c[r];
    }
  }
}

// ---------------------------------------------------------------------------
// Fused logits + online-softmax partials.
//   logits[n,v] = sum_k A[n,k] * W[v,k] + bias[v]     (W row-major [V,K])
// Block tile: 256 rows x 128 cols. 8 waves; each wave owns one 16-col strip
// and 16 row-tiles (16 v8f accumulators). Every b64 weight load feeds
// SIXTEEN independent v_wmma_f32_16x16x4_f32 chains -> weight bytes are
// fetched from L2 only N/256 = 4 times total (~600 MB vs 150 MB HBM).
// Emits per (colblock,row): (tile max, tile sum-exp); captures target-column
// logit (clipped index) and the head's 3 cluster logits (aux_out).
// ---------------------------------------------------------------------------
#define ROWS 256
#define KC   32

__global__ __launch_bounds__(256) void wmma_logits_kernel(
    const float* __restrict__ A, int K,
    const float* __restrict__ W, const float* __restrict__ Bias, int Vmain,
    const float* __restrict__ W2, const float* __restrict__ Bias2, int Vextra,
    const int* __restrict__ target, int tgt_off, int tgt_clipmax,
    float* __restrict__ partial,      // [CB][N][2]
    float* __restrict__ targ_out,     // [N]
    float* __restrict__ aux_out,      // [N*Vextra] or nullptr
    int N)
{
  const int Vtot = Vmain + Vextra;
  const int row0 = blockIdx.y * ROWS;
  const int wave = threadIdx.x >> 5;
  const int lane = threadIdx.x & 31;
  const int half = lane >> 4;
  const int ln   = lane & 15;
  const int col0 = blockIdx.x * 128 + wave * 16;
  const int col  = col0 + ln;
  const bool valid = (col < Vtot);

  // Stride 36: 16B-aligned b128 staging stores, 8B-aligned b64 fragment
  // loads, and 36*n mod 64 spans 16 distinct 4-bank groups -> conflict-free.
  __shared__ float As[ROWS][36];
  __shared__ int   st[ROWS];
  __shared__ float redM[8][ROWS];
  __shared__ float redS[8][ROWS];

  st[threadIdx.x] = target[row0 + threadIdx.x];

  // Per-lane weight row pointer + bias (invalid lanes read row 0 harmlessly).
  const float* wrow;
  float bias;
  if (col < Vmain)      { wrow = W  + (size_t)col * K;            bias = Bias[col]; }
  else if (valid)       { wrow = W2 + (size_t)(col - Vmain) * K;  bias = Bias2[col - Vmain]; }
  else                  { wrow = W;                               bias = 0.0f; }

  v8f acc[16] = {};
  for (int k0 = 0; k0 < K; k0 += KC) {
    const int kc = min(KC, K - k0);
    __syncthreads();
    {
      // One thread per row: up to 8 x global_load_b128 into the LDS tile.
      const float* Arow = A + (size_t)(row0 + threadIdx.x) * K + k0;
#pragma unroll
      for (int j = 0; j < KC / 4; ++j) {
        if (4 * j < kc)
          *(float4*)(&As[threadIdx.x][4 * j]) = *(const float4*)(Arow + 4 * j);
      }
    }
    __syncthreads();
#pragma unroll 2
    for (int kk = 0; kk < kc; kk += 4) {
      const int kg = kk + 2 * half;
      const v2f b = *(const v2f*)(wrow + k0 + kg);   // global_load_b64 weight stream
#pragma unroll
      for (int t = 0; t < 16; ++t) {
        v2f a = *(const v2f*)(&As[t * 16 + ln][kg]); // ds_load_b64
        acc[t] = WMMA_F32(a, b, acc[t]);             // rows row0+t*16 .. +15
      }
    }
  }

  // Epilogue: bias + masking, target / cluster capture, per-row max & sum-exp.
  // Row M of tile t lives in VGPR (M&7) across the 16 lanes of half (M>>3):
  // xor-shuffles with masks 1,2,4,8 stay inside each half on wave32.
  __syncthreads();
#pragma unroll
  for (int t = 0; t < 16; ++t) {
    float v[8], m[8], s[8];
#pragma unroll
    for (int r = 0; r < 8; ++r) {
      v[r] = valid ? (acc[t][r] + bias) : NEG_BIG;
      if (valid) {
        const int rl  = t * 16 + r + 8 * half;
        const int row = row0 + rl;
        const int own = min(max(st[rl] - tgt_off, 0), tgt_clipmax);
        if (own == col) targ_out[row] = v[r];
        if (aux_out && col >= Vmain) aux_out[(size_t)row * Vextra + (col - Vmain)] = v[r];
      }
      m[r] = v[r];
    }
#pragma unroll
    for (int mask = 1; mask <= 8; mask <<= 1)
#pragma unroll
      for (int r = 0; r < 8; ++r) m[r] = fmaxf(m[r], __shfl_xor(m[r], mask, 32));
#pragma unroll
    for (int r = 0; r < 8; ++r) s[r] = __expf(v[r] - m[r]);
#pragma unroll
    for (int mask = 1; mask <= 8; mask <<= 1)
#pragma unroll
      for (int r = 0; r < 8; ++r) s[r] += __shfl_xor(s[r], mask, 32);
    if (ln == 0) {
#pragma unroll
      for (int r = 0; r < 8; ++r) {
        const int rl = t * 16 + r + 8 * half;
        redM[wave][rl] = m[r];
        redS[wave][rl] = (m[r] <= NEG_THRESH) ? 0.0f : s[r];  // fully-masked tile
      }
    }
  }
  __syncthreads();
  {
    float M = NEG_BIG, S = 0.0f;
#pragma unroll
    for (int w = 0; w < 8; ++w) lse_combine(M, S, redM[w][threadIdx.x], redS[w][threadIdx.x]);
    const size_t idx = ((size_t)blockIdx.x * N + row0 + threadIdx.x) * 2;
    partial[idx + 0] = M;
    partial[idx + 1] = S;
  }
}

// ---------------------------------------------------------------------------
// Final per-row reduction: merge partials -> 4 log-sum-exps, assemble NLL.
// One block per row.
// ---------------------------------------------------------------------------
__global__ __launch_bounds__(256) void adaptive_nll_reduce_kernel(
    const float* __restrict__ pH, int cbH,
    const float* __restrict__ p1, int cb1,
    const float* __restrict__ p2, int cb2,
    const float* __restrict__ p3, int cb3,
    const float* __restrict__ tg0, const float* __restrict__ tg1,
    const float* __restrict__ tg2, const float* __restrict__ tg3,
    const float* __restrict__ clus, const int* __restrict__ target,
    float* __restrict__ out, int N)
{
  const int n = blockIdx.x;
  __shared__ float sM[256], sS[256];
  __shared__ float lse[4];

  const float* parr[4] = {pH, p1, p2, p3};
  const int    cbs[4]  = {cbH, cb1, cb2, cb3};

  for (int ci = 0; ci < 4; ++ci) {
    float M = NEG_BIG, S = 0.0f;
    const float* p = parr[ci];
    for (int cb = threadIdx.x; cb < cbs[ci]; cb += 256) {
      const size_t idx = ((size_t)cb * N + n) * 2;
      lse_combine(M, S, p[idx], p[idx + 1]);
    }
    sM[threadIdx.x] = M; sS[threadIdx.x] = S;
    for (int off = 128; off > 0; off >>= 1) {
      __syncthreads();
      if (threadIdx.x < off) {
        float Ma = sM[threadIdx.x], Sa = sS[threadIdx.x];
        lse_combine(Ma, Sa, sM[threadIdx.x + off], sS[threadIdx.x + off]);
        sM[threadIdx.x] = Ma; sS[threadIdx.x] = Sa;
      }
    }
    __syncthreads();
    if (threadIdx.x == 0) lse[ci] = sM[0] + __logf(sS[0]);
    __syncthreads();
  }

  if (threadIdx.x == 0) {
    const int t = target[n];
    float lp = tg0[n] - lse[0];                                   // shortlist
    if      (t >= 20000 && t <  40000) lp = (clus[n*3 + 2] - lse[0]) + (tg1[n] - lse[1]);
    else if (t >= 40000 && t < 200000) lp = (clus[n*3 + 1] - lse[0]) + (tg2[n] - lse[2]);
    else if (t >= 200000)              lp = (clus[n*3 + 0] - lse[0]) + (tg3[n] - lse[3]);
    out[n] = -lp;
  }
}

// ---------------------------------------------------------------------------
extern "C" void kernel_launch(void* const* d_in, const int* in_sizes, int n_in,
                              void* d_out, int out_size, void* d_ws, size_t ws_size,
                              hipStream_t stream) {
  const float* hidden    = (const float*)d_in[0];
  const int*   target    = (const int*)  d_in[1];
  const float* head_w    = (const float*)d_in[2];
  const float* head_b    = (const float*)d_in[3];
  const float* cluster_w = (const float*)d_in[4];
  const float* cluster_b = (const float*)d_in[5];
  const float* proj0     = (const float*)d_in[6];
  const float* proj1     = (const float*)d_in[7];
  const float* proj2     = (const float*)d_in[8];
  const float* proj3     = (const float*)d_in[9];
  const float* w1        = (const float*)d_in[10];
  const float* b1        = (const float*)d_in[11];
  const float* w2        = (const float*)d_in[12];
  const float* b2        = (const float*)d_in[13];
  const float* w3        = (const float*)d_in[14];
  const float* b3        = (const float*)d_in[15];
  float* out = (float*)d_out;

  const int N = in_sizes[1];     // 1024 tokens
  const int D = 1024;

  const int V0 = 20000, VC = 3;  // head + cluster
  const int V1 = 20000, V2 = 160000, V3 = 67735;
  const int CBH = (V0 + VC + 127) / 128;   // 157
  const int CB1 = (V1 + 127) / 128;        // 157
  const int CB2 = (V2 + 127) / 128;        // 1250
  const int CB3 = (V3 + 127) / 128;        // 530

  // Workspace carve-up (floats).
  float* ws = (float*)d_ws;
  size_t off = 0;
  float* h0  = ws + off; off += (size_t)N * 1024;
  float* h1  = ws + off; off += (size_t)N * 256;
  float* h2  = ws + off; off += (size_t)N * 64;
  float* h3  = ws + off; off += (size_t)N * 16;
  float* pH  = ws + off; off += (size_t)CBH * N * 2;
  float* p1  = ws + off; off += (size_t)CB1 * N * 2;
  float* p2  = ws + off; off += (size_t)CB2 * N * 2;
  float* p3  = ws + off; off += (size_t)CB3 * N * 2;
  float* tg0 = ws + off; off += (size_t)N;
  float* tg1 = ws + off; off += (size_t)N;
  float* tg2 = ws + off; off += (size_t)N;
  float* tg3 = ws + off; off += (size_t)N;
  float* clus = ws + off; off += (size_t)N * 3;
  (void)ws_size; (void)n_in; (void)out_size;

  const dim3 blk(256);
  const int RB16 = N / 16;
  const int RBL  = N / ROWS;    // 4 row-blocks for the fused logits kernels

  // Phase 1: projections h_i = hidden @ proj_i
  wmma_proj_kernel<<<dim3(1024/128, RB16), blk, 0, stream>>>(hidden, proj0, h0, N, D, 1024);
  wmma_proj_kernel<<<dim3( 256/128, RB16), blk, 0, stream>>>(hidden, proj1, h1, N, D, 256);
  wmma_proj_kernel<<<dim3(1,        RB16), blk, 0, stream>>>(hidden, proj2, h2, N, D, 64);
  wmma_proj_kernel<<<dim3(1,        RB16), blk, 0, stream>>>(hidden, proj3, h3, N, D, 16);

  // Phase 2: fused logits + online-softmax partials (single pass over weights).
  wmma_logits_kernel<<<dim3(CBH, RBL), blk, 0, stream>>>(
      h0, 1024, head_w, head_b, V0, cluster_w, cluster_b, VC,
      target, 0, V0 - 1, pH, tg0, clus, N);
  wmma_logits_kernel<<<dim3(CB1, RBL), blk, 0, stream>>>(
      h1, 256, w1, b1, V1, nullptr, nullptr, 0,
      target, 20000, V1 - 1, p1, tg1, nullptr, N);
  wmma_logits_kernel<<<dim3(CB2, RBL), blk, 0, stream>>>(
      h2, 64, w2, b2, V2, nullptr, nullptr, 0,
      target, 40000, V2 - 1, p2, tg2, nullptr, N);
  wmma_logits_kernel<<<dim3(CB3, RBL), blk, 0, stream>>>(
      h3, 16, w3, b3, V3, nullptr, nullptr, 0,
      target, 200000, V3 - 1, p3, tg3, nullptr, N);

  // Phase 3: per-row log-sum-exp merge + NLL assembly.
  adaptive_nll_reduce_kernel<<<dim3(N), blk, 0, stream>>>(
      pH, CBH, p1, CB1, p2, CB2, p3, CB3,
      tg0, tg1, tg2, tg3, clus, target, out, N);
}